// DeepClusteringLoss_15006615733128
// MI455X (gfx1250) — compile-verified
//
#include <hip/hip_runtime.h>

// Problem constants (from reference)
#define BATCH 8
#define FREQ  257
#define TIMEL 1000
#define EMB   20
#define NSRC  2
#define NROWS (FREQ * TIMEL)          // 257000 rows per batch

// Parallelization
#define WAVES_PER_BATCH 128
#define THREADS 256
#define WAVES_PER_BLOCK (THREADS / 32)
#define ROWS_PER_WAVE 2008            // ceil(257000/128); 2008 % 8 == 0, tail slab 1984 % 8 == 0

// Workspace: per batch 3 tiles of 16x16 f32 = 768 floats
#define TILE_FLOATS 768

typedef __attribute__((ext_vector_type(2))) float v2f;
typedef __attribute__((ext_vector_type(8))) float v8f;

// ---------------------------------------------------------------------------
// Zero the accumulator workspace (atomics accumulate into it each launch).
// ---------------------------------------------------------------------------
__global__ void dc_zero_kernel(float* __restrict__ p, int n) {
    int i = blockIdx.x * blockDim.x + threadIdx.x;
    if (i < n) p[i] = 0.0f;
}

// ---------------------------------------------------------------------------
// Main pass: per-batch Gram matrix of the padded row W = [E(20) | V(2) | 1 | 0..]
// via V_WMMA_F32_16X16X4_F32 rank-4 updates. WtW is symmetric -> 3 tiles:
//   t00 = cols 0-15 x 0-15, t01 = cols 0-15 x 16-31, t11 = 16-31 x 16-31.
//
// Per-lane operand construction (ISA 7.12.2, 32-bit 16x4 A layout):
//   lane L: m = L&15 (tile row/col), kb = (L>>4)*2 (k of .x; .y is k+1)
// The 4x16 B operand uses the same k-striping, so for the symmetric product the
// A-register and B-register contents coincide.
//
// Hot loop is fully branch-free: the per-lane tile-1 column source (E / V /
// ones / pad) is hoisted into a load recipe {base ptr, stride, mul, add} so
// every lane issues plain unconditional loads; row predication exists only in
// a cold tail (never taken for these constants since all slabs are 8-aligned).
// ---------------------------------------------------------------------------
__global__ __launch_bounds__(THREADS)
void dc_gram_kernel(const float* __restrict__ E,
                    const float* __restrict__ V,
                    float* __restrict__ acc) {
    const int lane = threadIdx.x & 31;
    const int wid  = blockIdx.x * WAVES_PER_BLOCK + (threadIdx.x >> 5);
    const int b    = wid / WAVES_PER_BATCH;
    const int slab = wid % WAVES_PER_BATCH;

    const long long row0 = (long long)slab * ROWS_PER_WAVE;
    long long rowEnd = row0 + ROWS_PER_WAVE;
    if (rowEnd > NROWS) rowEnd = NROWS;

    const float* __restrict__ Eb = E + (long long)b * NROWS * EMB;
    const float* __restrict__ Vb = V + (long long)b * NROWS * NSRC;

    const int m  = lane & 15;          // row/col within tile
    const int kb = (lane >> 4) << 1;   // k base: lanes 0-15 -> k0/k1, 16-31 -> k2/k3
    const int c1 = 16 + m;             // padded-W column for tile 1

    // ---- loop-invariant load recipe for tile-1 column ----
    const float* p1base;
    int   st1;                // element stride between consecutive rows
    float mul1, add1;         // value = mul1 * load + add1
    if (c1 < EMB) {                     // E columns 16..19
        p1base = Eb + c1;  st1 = EMB;  mul1 = 1.0f; add1 = 0.0f;
    } else if (c1 < EMB + NSRC) {       // V columns 20..21
        p1base = Vb + (c1 - EMB); st1 = NSRC; mul1 = 1.0f; add1 = 0.0f;
    } else if (c1 == EMB + NSRC) {      // ones column 22
        p1base = Eb; st1 = 0; mul1 = 0.0f; add1 = 1.0f;
    } else {                            // pad columns 23..31
        p1base = Eb; st1 = 0; mul1 = 0.0f; add1 = 0.0f;
    }
    const int st1x4 = st1 * 4;
    const int st1x8 = st1 * 8;

    v8f a00 = {}, a01 = {}, a11 = {};
    v8f b00 = {}, b01 = {}, b11 = {};

    // Running 32-bit offsets (max ~5.1M elements, fits easily).
    unsigned offA  = (unsigned)((row0 + kb) * EMB + m);   // E, row n0, col m
    unsigned offB0 = (unsigned)((row0 + kb) * st1);       // tile1, row n0
    unsigned offB1 = offB0 + (unsigned)st1;               // tile1, row n0+1

    const long long total = rowEnd - row0;
    const long long nFull = total & ~7LL;                 // 8-row super-chunks

    for (long long i = 0; i < nFull; i += 8) {
        v2f p0a, p1a, p0b, p1b;
        // chunk A: rows i .. i+3
        p0a.x = Eb[offA];
        p0a.y = Eb[offA + EMB];
        p1a.x = fmaf(mul1, p1base[offB0], add1);
        p1a.y = fmaf(mul1, p1base[offB1], add1);
        // chunk B: rows i+4 .. i+7
        p0b.x = Eb[offA + 4 * EMB];
        p0b.y = Eb[offA + 5 * EMB];
        p1b.x = fmaf(mul1, p1base[offB0 + st1x4], add1);
        p1b.y = fmaf(mul1, p1base[offB1 + st1x4], add1);

        a00 = __builtin_amdgcn_wmma_f32_16x16x4_f32(false, p0a, false, p0a,
                                                    (short)0, a00, false, false);
        a01 = __builtin_amdgcn_wmma_f32_16x16x4_f32(false, p0a, false, p1a,
                                                    (short)0, a01, false, false);
        a11 = __builtin_amdgcn_wmma_f32_16x16x4_f32(false, p1a, false, p1a,
                                                    (short)0, a11, false, false);
        b00 = __builtin_amdgcn_wmma_f32_16x16x4_f32(false, p0b, false, p0b,
                                                    (short)0, b00, false, false);
        b01 = __builtin_amdgcn_wmma_f32_16x16x4_f32(false, p0b, false, p1b,
                                                    (short)0, b01, false, false);
        b11 = __builtin_amdgcn_wmma_f32_16x16x4_f32(false, p1b, false, p1b,
                                                    (short)0, b11, false, false);

        offA  += 8 * EMB;
        offB0 += st1x8;
        offB1 += st1x8;
    }

    // Cold predicated tail (not taken for current constants; kept for safety).
    for (long long r = row0 + nFull; r < rowEnd; r += 4) {
        const long long n0 = r + kb;
        const long long n1 = n0 + 1;
        const bool ok0 = (n0 < rowEnd);
        const bool ok1 = (n1 < rowEnd);
        v2f p0, p1;
        p0.x = ok0 ? Eb[n0 * EMB + m] : 0.0f;
        p0.y = ok1 ? Eb[n1 * EMB + m] : 0.0f;
        p1.x = ok0 ? fmaf(mul1, p1base[n0 * st1], add1) : 0.0f;
        p1.y = ok1 ? fmaf(mul1, p1base[n1 * st1], add1) : 0.0f;
        a00 = __builtin_amdgcn_wmma_f32_16x16x4_f32(false, p0, false, p0,
                                                    (short)0, a00, false, false);
        a01 = __builtin_amdgcn_wmma_f32_16x16x4_f32(false, p0, false, p1,
                                                    (short)0, a01, false, false);
        a11 = __builtin_amdgcn_wmma_f32_16x16x4_f32(false, p1, false, p1,
                                                    (short)0, a11, false, false);
    }

    // Merge the two accumulator sets.
#pragma unroll
    for (int r = 0; r < 8; ++r) {
        a00[r] += b00[r];
        a01[r] += b01[r];
        a11[r] += b11[r];
    }

    // C/D layout (ISA 7.12.2): VGPR reg, lane L -> M = reg + (L>>4)*8, N = L&15.
    float* t00 = acc + (long long)b * TILE_FLOATS;
    float* t01 = t00 + 256;
    float* t11 = t00 + 512;
    const int half = lane >> 4;
    const int ncol = lane & 15;
#pragma unroll
    for (int reg = 0; reg < 8; ++reg) {
        const int idx = (reg + half * 8) * 16 + ncol;
        atomicAdd(&t00[idx], a00[reg]);
        atomicAdd(&t01[idx], a01[reg]);
        atomicAdd(&t11[idx], a11[reg]);
    }
}

// ---------------------------------------------------------------------------
// Finalize: reconstruct G (20x20), EtV (20x2), sumV (2) from the 3 symmetric
// tiles; loss = sum((EtV/(sumV+eps))^2), norm = mean_b ||G||_F^2.
// Single wave; shfl reduction.
// ---------------------------------------------------------------------------
__device__ __forceinline__ float dc_elem(const float* t00, const float* t01,
                                         const float* t11, int i, int j) {
    if (i < 16 && j < 16) return t00[i * 16 + j];
    if (i < 16)           return t01[i * 16 + (j - 16)];
    if (j < 16)           return t01[j * 16 + (i - 16)];   // symmetric
    return t11[(i - 16) * 16 + (j - 16)];
}

__global__ void dc_finalize_kernel(const float* __restrict__ acc,
                                   float* __restrict__ out) {
    const int lane = threadIdx.x;  // launched with 32 threads
    float loss = 0.0f;
    float norm = 0.0f;

    for (int b = 0; b < BATCH; ++b) {
        const float* t00 = acc + (long long)b * TILE_FLOATS;
        const float* t01 = t00 + 256;
        const float* t11 = t00 + 512;
        const float sv0 = dc_elem(t00, t01, t11, 22, 20) + 1e-8f;
        const float sv1 = dc_elem(t00, t01, t11, 22, 21) + 1e-8f;

        // Flat work: 400 Gram elements + 40 EtY elements = 440.
        for (int idx = lane; idx < 440; idx += 32) {
            if (idx < 400) {
                const int d = idx / 20;
                const int e = idx % 20;
                const float g = dc_elem(t00, t01, t11, d, e);
                norm += g * g;
            } else {
                const int t = idx - 400;
                const int d = t >> 1;
                const int s = t & 1;
                const float ety =
                    dc_elem(t00, t01, t11, d, 20 + s) / (s ? sv1 : sv0);
                loss += ety * ety;
            }
        }
    }

    for (int off = 16; off > 0; off >>= 1) {
        loss += __shfl_xor(loss, off, 32);
        norm += __shfl_xor(norm, off, 32);
    }

    if (lane == 0) {
        const float norm_term = norm / (float)BATCH;
        out[0] = -loss / (norm_term + 1e-8f);
    }
}

// ---------------------------------------------------------------------------
extern "C" void kernel_launch(void* const* d_in, const int* in_sizes, int n_in,
                              void* d_out, int out_size, void* d_ws, size_t ws_size,
                              hipStream_t stream) {
    const float* E = (const float*)d_in[0];   // (8,257,1000,20) f32
    const float* V = (const float*)d_in[1];   // (8,257,1000,2)  f32
    float* acc = (float*)d_ws;                // needs 8*768*4 = 24576 bytes
    float* out = (float*)d_out;

    const int accN = BATCH * TILE_FLOATS;
    dc_zero_kernel<<<(accN + 255) / 256, 256, 0, stream>>>(acc, accN);

    const int nBlocks = (BATCH * WAVES_PER_BATCH) / WAVES_PER_BLOCK;  // 128
    dc_gram_kernel<<<nBlocks, THREADS, 0, stream>>>(E, V, acc);

    dc_finalize_kernel<<<1, 32, 0, stream>>>(acc, out);
}